// HRM_MC_E3_83399674954323
// MI455X (gfx1250) — compile-verified
//
#include <hip/hip_runtime.h>
#include <math.h>

// Problem constants (match reference)
#define BB   8
#define NN   4096
#define EE   131072
#define DIMD 64
#define DIMH 128

typedef float v2f __attribute__((ext_vector_type(2)));
typedef float v8f __attribute__((ext_vector_type(8)));

__device__ __forceinline__ float silu_f(float x) {
    return x * (1.0f / (1.0f + __expf(-x)));
}

// D = A(16x4) * B(4x16) + C, fp32 WMMA
__device__ __forceinline__ v8f wmma4(v2f a, v2f b, v8f c) {
    return __builtin_amdgcn_wmma_f32_16x16x4_f32(
        false, a, false, b, (short)0, c, false, false);
}

// ---------------------------------------------------------------------------
// Zero workspace (agg [B,N,H] and coord_delta [B,N,3], contiguous)
// ---------------------------------------------------------------------------
__global__ void egnn_zero_ws(float* __restrict__ p, size_t n) {
    size_t i = (size_t)blockIdx.x * blockDim.x + threadIdx.x;
    size_t stride = (size_t)gridDim.x * blockDim.x;
    for (; i < n; i += stride) p[i] = 0.0f;
}

// ---------------------------------------------------------------------------
// Edge kernel: one wave = one 16-edge tile.
//   h1  = silu([nf_src | nf_dst | dist] @ Wm1 + bm1)   (K=129, dist folded)
//   msg = h1 @ Wm2 + bm2
//   agg[dst]        += msg          (atomics)
//   ch  = silu(msg @ Wc1 + bc1); cw = ch @ Wc2 + bc2
//   cdelta[dst]     += cw * unit_vec(dst - src)        (atomics)
// 4 waves / block, per-wave LDS: h1(16x128) + msg(16x128) = 16 KB -> 64 KB dyn
// ---------------------------------------------------------------------------
__global__ void __launch_bounds__(128)
egnn_edge(const float* __restrict__ nf, const float* __restrict__ coords,
          const int* __restrict__ eidx,
          const float* __restrict__ Wm1, const float* __restrict__ bm1,
          const float* __restrict__ Wm2, const float* __restrict__ bm2,
          const float* __restrict__ Wc1, const float* __restrict__ bc1,
          const float* __restrict__ Wc2, const float* __restrict__ bc2,
          float* __restrict__ agg, float* __restrict__ cdelta)
{
    extern __shared__ float smem[];
    const int lane = threadIdx.x & 31;
    const int warp = threadIdx.x >> 5;
    float* hbuf = smem + warp * 4096;   // 16 x 128 : h1, later coord-hidden
    float* mbuf = hbuf + 2048;          // 16 x 128 : messages

    const int tid = blockIdx.x * 4 + warp;        // 16-edge tile id
    const int tilesPerBatch = EE / 16;
    const int b  = tid / tilesPerBatch;
    const int e0 = (tid % tilesPerBatch) * 16;
    const size_t nfb = (size_t)b * NN * DIMD;

    // lanes 0..15 own edge e0+lane: indices + geometry
    int src = 0, dst = 0;
    float dist = 0.f, ux = 0.f, uy = 0.f, uz = 0.f;
    if (lane < 16) {
        src = eidx[e0 + lane];
        dst = eidx[EE + e0 + lane];
        const float* cs = coords + ((size_t)b * NN + src) * 3;
        const float* cd = coords + ((size_t)b * NN + dst) * 3;
        float dx = cd[0] - cs[0], dy = cd[1] - cs[1], dz = cd[2] - cs[2];
        dist = sqrtf(dx * dx + dy * dy + dz * dz);
        float inv = 1.0f / (dist + 1e-8f);
        ux = dx * inv; uy = dy * inv; uz = dz * inv;
    }

    const int m     = lane & 15;          // A row / B-C col for this lane
    const int koff  = (lane >> 4) << 1;   // 0 (lanes 0-15) or 2 (lanes 16-31)
    const int rbase = (lane >> 4) * 8;    // C/D rows rbase..rbase+7

    // row-m edge endpoints (needed by all 32 lanes for A-fragment loads)
    const int msrc = __shfl(src, m);
    const int mdst = __shfl(dst, m);
    const float* arow_s = nf + nfb + (size_t)msrc * DIMD;
    const float* arow_d = nf + nfb + (size_t)mdst * DIMD;

    // dist of each accumulator row (rank-1 fold of the 129th input column)
    float distr[8];
    #pragma unroll
    for (int r = 0; r < 8; ++r) distr[r] = __shfl(dist, rbase + r);

    // ---- message layer 1 -> hbuf (silu applied) ----
    #pragma unroll 1
    for (int nt = 0; nt < 8; ++nt) {
        const int col = nt * 16 + m;
        const float b1   = bm1[col];
        const float w128 = Wm1[128 * DIMH + col];     // dist row of Wm1
        v8f acc;
        #pragma unroll
        for (int r = 0; r < 8; ++r) acc[r] = b1 + distr[r] * w128;
        #pragma unroll
        for (int k0 = 0; k0 < 128; k0 += 4) {
            const float* ar = (k0 < 64) ? (arow_s + k0) : (arow_d + (k0 - 64));
            v2f a, w;
            a.x = ar[koff];
            a.y = ar[koff + 1];
            w.x = Wm1[(size_t)(k0 + koff)     * DIMH + col];
            w.y = Wm1[(size_t)(k0 + koff + 1) * DIMH + col];
            acc = wmma4(a, w, acc);
        }
        #pragma unroll
        for (int r = 0; r < 8; ++r)
            hbuf[(rbase + r) * DIMH + col] = silu_f(acc[r]);
    }

    // ---- message layer 2 -> mbuf (no activation) ----
    #pragma unroll 1
    for (int nt = 0; nt < 8; ++nt) {
        const int col = nt * 16 + m;
        const float b2 = bm2[col];
        v8f acc = {b2, b2, b2, b2, b2, b2, b2, b2};
        #pragma unroll
        for (int k0 = 0; k0 < 128; k0 += 4) {
            v2f a, w;
            a.x = hbuf[m * DIMH + k0 + koff];
            a.y = hbuf[m * DIMH + k0 + koff + 1];
            w.x = Wm2[(size_t)(k0 + koff)     * DIMH + col];
            w.y = Wm2[(size_t)(k0 + koff + 1) * DIMH + col];
            acc = wmma4(a, w, acc);
        }
        #pragma unroll
        for (int r = 0; r < 8; ++r)
            mbuf[(rbase + r) * DIMH + col] = acc[r];
    }

    // ---- coord layer 1: silu(msg @ Wc1 + bc1) -> hbuf (h1 is dead) ----
    #pragma unroll 1
    for (int nt = 0; nt < 8; ++nt) {
        const int col = nt * 16 + m;
        const float bc = bc1[col];
        v8f acc = {bc, bc, bc, bc, bc, bc, bc, bc};
        #pragma unroll
        for (int k0 = 0; k0 < 128; k0 += 4) {
            v2f a, w;
            a.x = mbuf[m * DIMH + k0 + koff];
            a.y = mbuf[m * DIMH + k0 + koff + 1];
            w.x = Wc1[(size_t)(k0 + koff)     * DIMH + col];
            w.y = Wc1[(size_t)(k0 + koff + 1) * DIMH + col];
            acc = wmma4(a, w, acc);
        }
        #pragma unroll
        for (int r = 0; r < 8; ++r)
            hbuf[(rbase + r) * DIMH + col] = silu_f(acc[r]);
    }

    // ---- coord weight (output dim 1) + equivariant coord scatter ----
    if (lane < 16) {
        float s = bc2[0];
        #pragma unroll 4
        for (int k = 0; k < DIMH; ++k)
            s += hbuf[lane * DIMH + k] * Wc2[k];
        float* cd = cdelta + ((size_t)b * NN + dst) * 3;
        atomicAdd(cd + 0, s * ux);
        atomicAdd(cd + 1, s * uy);
        atomicAdd(cd + 2, s * uz);
    }

    // ---- scatter-add messages into agg[b, dst, :] ----
    for (int i = lane; i < 16 * DIMH; i += 32) {
        const int e = i >> 7;          // edge within tile
        const int h = i & (DIMH - 1);
        const int d = __shfl(dst, e);
        atomicAdd(agg + ((size_t)b * NN + d) * DIMH + h, mbuf[e * DIMH + h]);
    }
}

// ---------------------------------------------------------------------------
// Node kernel: one wave = one 16-node tile.
//   h  = silu([nf | agg] @ Wn1 + bn1)   (K=192)
//   nf_new = nf + h @ Wn2 + bn2
//   coords_new = coords + cdelta
// 8 waves / block, per-wave LDS: h(16x128) = 8 KB -> 64 KB dyn
// ---------------------------------------------------------------------------
__global__ void __launch_bounds__(256)
egnn_node(const float* __restrict__ nf, const float* __restrict__ coords,
          const float* __restrict__ agg, const float* __restrict__ cdelta,
          const float* __restrict__ Wn1, const float* __restrict__ bn1,
          const float* __restrict__ Wn2, const float* __restrict__ bn2,
          float* __restrict__ out_nf, float* __restrict__ out_coords)
{
    extern __shared__ float smem[];
    const int lane = threadIdx.x & 31;
    const int warp = threadIdx.x >> 5;
    float* hbuf = smem + warp * 2048;   // 16 x 128

    const int tid = blockIdx.x * 8 + warp;        // 16-node tile id
    const int tilesPerBatch = NN / 16;
    const int b  = tid / tilesPerBatch;
    const int t0 = (tid % tilesPerBatch) * 16;

    const int m     = lane & 15;
    const int koff  = (lane >> 4) << 1;
    const int rbase = (lane >> 4) * 8;

    const size_t nodeM = (size_t)b * NN + t0 + m;
    const float* anf = nf  + nodeM * DIMD;
    const float* aag = agg + nodeM * DIMH;

    // ---- layer 1: K = 192 ([nf | agg]) -> hbuf (silu) ----
    #pragma unroll 1
    for (int nt = 0; nt < 8; ++nt) {
        const int col = nt * 16 + m;
        const float b1 = bn1[col];
        v8f acc = {b1, b1, b1, b1, b1, b1, b1, b1};
        #pragma unroll
        for (int k0 = 0; k0 < 192; k0 += 4) {
            const float* ar = (k0 < 64) ? (anf + k0) : (aag + (k0 - 64));
            v2f a, w;
            a.x = ar[koff];
            a.y = ar[koff + 1];
            w.x = Wn1[(size_t)(k0 + koff)     * DIMH + col];
            w.y = Wn1[(size_t)(k0 + koff + 1) * DIMH + col];
            acc = wmma4(a, w, acc);
        }
        #pragma unroll
        for (int r = 0; r < 8; ++r)
            hbuf[(rbase + r) * DIMH + col] = silu_f(acc[r]);
    }

    // ---- layer 2: K = 128 -> 64 cols, residual add, store global ----
    #pragma unroll 1
    for (int nt = 0; nt < 4; ++nt) {
        const int col = nt * 16 + m;
        const float b2 = bn2[col];
        v8f acc = {b2, b2, b2, b2, b2, b2, b2, b2};
        #pragma unroll
        for (int k0 = 0; k0 < 128; k0 += 4) {
            v2f a, w;
            a.x = hbuf[m * DIMH + k0 + koff];
            a.y = hbuf[m * DIMH + k0 + koff + 1];
            w.x = Wn2[(size_t)(k0 + koff)     * DIMD + col];
            w.y = Wn2[(size_t)(k0 + koff + 1) * DIMD + col];
            acc = wmma4(a, w, acc);
        }
        #pragma unroll
        for (int r = 0; r < 8; ++r) {
            const size_t node = (size_t)b * NN + t0 + rbase + r;
            out_nf[node * DIMD + col] = nf[node * DIMD + col] + acc[r];
        }
    }

    // ---- coords_new = coords + cdelta ----
    if (lane < 16) {
        const size_t node = (size_t)b * NN + t0 + lane;
        #pragma unroll
        for (int j = 0; j < 3; ++j)
            out_coords[node * 3 + j] = coords[node * 3 + j] + cdelta[node * 3 + j];
    }
}

// ---------------------------------------------------------------------------
extern "C" void kernel_launch(void* const* d_in, const int* in_sizes, int n_in,
                              void* d_out, int out_size, void* d_ws, size_t ws_size,
                              hipStream_t stream)
{
    const float* nf     = (const float*)d_in[0];
    const float* coords = (const float*)d_in[1];
    const int*   eidx   = (const int*)  d_in[2];
    const float* Wm1 = (const float*)d_in[3];
    const float* bm1 = (const float*)d_in[4];
    const float* Wm2 = (const float*)d_in[5];
    const float* bm2 = (const float*)d_in[6];
    const float* Wn1 = (const float*)d_in[7];
    const float* bn1 = (const float*)d_in[8];
    const float* Wn2 = (const float*)d_in[9];
    const float* bn2 = (const float*)d_in[10];
    const float* Wc1 = (const float*)d_in[11];
    const float* bc1 = (const float*)d_in[12];
    const float* Wc2 = (const float*)d_in[13];
    const float* bc2 = (const float*)d_in[14];

    float* out_nf     = (float*)d_out;
    float* out_coords = out_nf + (size_t)BB * NN * DIMD;

    float* agg    = (float*)d_ws;                       // [B,N,H]
    float* cdelta = agg + (size_t)BB * NN * DIMH;       // [B,N,3]

    // zero scratch every call (harness poisons ws, does not re-zero)
    const size_t zn = (size_t)BB * NN * DIMH + (size_t)BB * NN * 3;
    egnn_zero_ws<<<1024, 256, 0, stream>>>(agg, zn);

    // edge pass: B*E/16 wave-tiles, 4 waves per block, 64 KB dynamic LDS
    const int edgeBlocks = (BB * (EE / 16)) / 4;        // 16384
    egnn_edge<<<edgeBlocks, 128, 4 * 4096 * sizeof(float), stream>>>(
        nf, coords, eidx, Wm1, bm1, Wm2, bm2, Wc1, bc1, Wc2, bc2, agg, cdelta);

    // node pass: B*N/16 wave-tiles, 8 waves per block, 64 KB dynamic LDS
    const int nodeBlocks = (BB * (NN / 16)) / 8;        // 256
    egnn_node<<<nodeBlocks, 256, 8 * 2048 * sizeof(float), stream>>>(
        nf, coords, agg, cdelta, Wn1, bn1, Wn2, bn2, out_nf, out_coords);
}